// AV_HAN_41704132445076
// MI455X (gfx1250) — compile-verified
//
#include <hip/hip_runtime.h>
#include <hip/hip_bf16.h>

typedef __attribute__((ext_vector_type(2))) float v2f;
typedef __attribute__((ext_vector_type(8))) float v8f;

#define BB   128
#define NT   560
#define NI   512
#define NA   48
#define DD   192
#define NEDGE 4096
#define SLOPE 0.2f

// ---------------- zero adjacency accumulators ----------------
__global__ void k_zero(unsigned* __restrict__ p, int nwords) {
    int i = blockIdx.x * blockDim.x + threadIdx.x;
    if (i < nwords) p[i] = 0u;
}

// ---------------- build bitmask adjacency ----------------
// I2A_row[b][u]      : u64, bits over aud a (edge img u -> aud a)
// A2I_row[b][k][16]  : u32 words, bits over img v (edge aud k -> img v)
// I2A_colT[b][a][16] : u32 words, bits over img u (edge img u -> aud a)
__global__ void k_build_adj(const int* __restrict__ E,
                            unsigned* __restrict__ i2a_row,   // as u32, 2 words/row
                            unsigned* __restrict__ a2i_row,
                            unsigned* __restrict__ i2a_colT) {
    int b = blockIdx.x;
    const int* Edst = E + (size_t)b * NEDGE;                 // channel 0
    const int* Esrc = E + (size_t)BB * NEDGE + (size_t)b * NEDGE; // channel 1
    for (int k = threadIdx.x; k < NEDGE; k += blockDim.x) {
        int dst = Edst[k];
        int src = Esrc[k];
        if (src >= 0 && src < NI && dst >= NI && dst < NT) {
            int a = dst - NI;
            atomicOr(&i2a_row[((size_t)b * NI + src) * 2 + (a >> 5)], 1u << (a & 31));
            atomicOr(&i2a_colT[((size_t)b * NA + a) * 16 + (src >> 5)], 1u << (src & 31));
        }
        if (src >= NI && src < NT && dst >= 0 && dst < NI) {
            int a = src - NI;
            atomicOr(&a2i_row[((size_t)b * NA + a) * 16 + (dst >> 5)], 1u << (dst & 31));
        }
    }
    if (threadIdx.x == 0) {
        // forced A_i2a[511, 47] = 1
        atomicOr(&i2a_row[((size_t)b * NI + (NI - 1)) * 2 + 1], 1u << 15);
        atomicOr(&i2a_colT[((size_t)b * NA + (NA - 1)) * 16 + 15], 1u << 31);
    }
}

// ---------------- M_img transposed rows: MimgT[b][v] bits over sources u ----------------
__global__ void k_mimgT(const unsigned* __restrict__ a2i_row,
                        const unsigned* __restrict__ i2a_colT,
                        unsigned* __restrict__ mimgT) {
    int t = blockIdx.x * blockDim.x + threadIdx.x;       // B*512*16 threads
    int b = t >> 13;            // /(512*16)
    int r = t & 8191;
    int v = r >> 4;
    int w = r & 15;
    unsigned acc = 0;
    int vw = v >> 5, vb = v & 31;
    const unsigned* arow = a2i_row + (size_t)b * NA * 16;
    const unsigned* ct   = i2a_colT + (size_t)b * NA * 16;
    #pragma unroll 4
    for (int k = 0; k < NA; ++k) {
        if ((arow[k * 16 + vw] >> vb) & 1u) acc |= ct[k * 16 + w];
    }
    mimgT[((size_t)b * NI + v) * 16 + w] = acc;
}

// ---------------- M_aud transposed rows: MaudT[b][v] (u64 as 2 u32) over sources u ----------------
__global__ void k_maudT(const unsigned* __restrict__ a2i_row,
                        const unsigned* __restrict__ i2a_colT,
                        unsigned* __restrict__ maudT) {
    int t = blockIdx.x * blockDim.x + threadIdx.x;       // B*48 threads
    int b = t / NA;
    int v = t % NA;
    unsigned lo = 0, hi = 0;
    const unsigned* cv = i2a_colT + ((size_t)b * NA + v) * 16;
    for (int u = 0; u < NA; ++u) {
        const unsigned* au = a2i_row + ((size_t)b * NA + u) * 16;
        unsigned any = 0;
        #pragma unroll
        for (int w = 0; w < 16; ++w) any |= au[w] & cv[w];
        if (any) { if (u < 32) lo |= 1u << u; else hi |= 1u << (u - 32); }
    }
    maudT[((size_t)b * NA + v) * 2 + 0] = lo;
    maudT[((size_t)b * NA + v) * 2 + 1] = hi;
}

// ---------------- fp32 WMMA GEMM: Y[b, nodebase+m, o] = sum_d H[b, nodebase+m, d] * W[o, d] (+bias) ----
// one wave computes a 16x16 tile; K=192 via 48x V_WMMA_F32_16X16X4_F32
__global__ void k_gemm_wmma(const float* __restrict__ H, const float* __restrict__ W,
                            const float* __restrict__ bias, float* __restrict__ Y,
                            int nodebase, int mtiles) {
    int wid  = blockIdx.x * (blockDim.x >> 5) + (threadIdx.x >> 5);
    int lane = threadIdx.x & 31;
    int tilesPerB = mtiles * (DD / 16);
    int b  = wid / tilesPerB;
    int r  = wid - b * tilesPerB;
    int mt = r / (DD / 16);
    int nt = r - mt * (DD / 16);

    int rl   = lane & 15;          // row within A-tile / col within B-tile
    int koff = (lane >> 4) << 1;   // lanes 16-31 handle K+2,K+3
    const float* hrow = H + ((size_t)b * NT + nodebase + mt * 16 + rl) * DD + koff;
    const float* wrow = W + (size_t)(nt * 16 + rl) * DD + koff;

    v8f acc = {};
    #pragma unroll 4
    for (int kk = 0; kk < DD; kk += 4) {
        v2f a, bb;
        a.x  = hrow[kk];  a.y  = hrow[kk + 1];
        bb.x = wrow[kk];  bb.y = wrow[kk + 1];
        acc = __builtin_amdgcn_wmma_f32_16x16x4_f32(false, a, false, bb,
                                                    (short)0, acc, false, false);
    }

    float bv = bias ? bias[nt * 16 + rl] : 0.0f;
    int mbase = (lane >> 4) << 3;  // lanes 16-31 hold rows M+8
    float* yrow = Y + ((size_t)b * NT + nodebase + mt * 16) * DD + nt * 16 + rl;
    #pragma unroll
    for (int j = 0; j < 8; ++j) {
        yrow[(size_t)(mbase + j) * DD] = acc[j] + bv;
    }
}

// ---------------- attention logits el/er per node ----------------
__global__ void k_elr(const float* __restrict__ Wh,
                      const float* __restrict__ ali, const float* __restrict__ ari,
                      const float* __restrict__ ala, const float* __restrict__ ara,
                      float* __restrict__ el, float* __restrict__ er) {
    int t = blockIdx.x * blockDim.x + threadIdx.x;       // B*NT threads
    int node = t % NT;
    const float* al = (node < NI) ? ali : ala;
    const float* ar = (node < NI) ? ari : ara;
    const float* w = Wh + (size_t)t * DD;
    float s1 = 0.f, s2 = 0.f;
    #pragma unroll 4
    for (int o = 0; o < DD; ++o) { float x = w[o]; s1 += x * al[o]; s2 += x * ar[o]; }
    el[t] = s1; er[t] = s2;
}

// ---------------- masked-softmax aggregation + bias + elu (one wave per destination) -----
__global__ void k_gat(const unsigned* __restrict__ maskrows, int nwords, int rowstride,
                      const float* __restrict__ Wh, const float* __restrict__ el,
                      const float* __restrict__ er, const float* __restrict__ bias,
                      float* __restrict__ gat, int nodebase, int ndst) {
    int wid  = blockIdx.x * (blockDim.x >> 5) + (threadIdx.x >> 5);
    int lane = threadIdx.x & 31;
    int b = wid / ndst;
    int v = wid - b * ndst;
    const unsigned* mrow = maskrows + (size_t)(b * ndst + v) * rowstride;
    const float* elb = el + (size_t)b * NT + nodebase;
    float erv = er[(size_t)b * NT + nodebase + v];

    // pass 1: max over adjacent sources
    float mx = -3.4e38f;
    bool any = false;
    for (int w = 0; w < nwords; ++w) {
        unsigned m = mrow[w];
        while (m) {
            int u = (w << 5) + __builtin_ctz(m); m &= m - 1;
            float e = elb[u] + erv;
            e = (e > 0.f) ? e : SLOPE * e;
            mx = fmaxf(mx, e);
            any = true;
        }
    }

    float acc[DD / 32];
    #pragma unroll
    for (int j = 0; j < DD / 32; ++j) acc[j] = 0.f;

    if (any) {
        // pass 2: sum of exp
        float s = 0.f;
        for (int w = 0; w < nwords; ++w) {
            unsigned m = mrow[w];
            while (m) {
                int u = (w << 5) + __builtin_ctz(m); m &= m - 1;
                float e = elb[u] + erv;
                e = (e > 0.f) ? e : SLOPE * e;
                s += expf(e - mx);
            }
        }
        float inv = 1.0f / s;
        // pass 3: weighted feature accumulation
        for (int w = 0; w < nwords; ++w) {
            unsigned m = mrow[w];
            while (m) {
                int u = (w << 5) + __builtin_ctz(m); m &= m - 1;
                float e = elb[u] + erv;
                e = (e > 0.f) ? e : SLOPE * e;
                float alpha = expf(e - mx) * inv;
                const float* wh = Wh + ((size_t)b * NT + nodebase + u) * DD;
                #pragma unroll
                for (int j = 0; j < DD / 32; ++j)
                    acc[j] += alpha * wh[j * 32 + lane];
            }
        }
    }

    float* out = gat + ((size_t)b * NT + nodebase + v) * DD;
    #pragma unroll
    for (int j = 0; j < DD / 32; ++j) {
        int o = j * 32 + lane;
        float x = acc[j] + bias[o];
        out[o] = (x > 0.f) ? x : expm1f(x);   // elu
    }
}

// ---------------- layernorm over D, in-place on output (one wave per node) -------------
__global__ void k_layernorm(float* __restrict__ Y,
                            const float* __restrict__ g1, const float* __restrict__ b1,
                            const float* __restrict__ g2, const float* __restrict__ b2) {
    int wid  = blockIdx.x * (blockDim.x >> 5) + (threadIdx.x >> 5);  // B*NT waves
    int lane = threadIdx.x & 31;
    int node = wid % NT;
    float* row = Y + (size_t)wid * DD;

    float x[DD / 32];
    float s = 0.f, sq = 0.f;
    #pragma unroll
    for (int j = 0; j < DD / 32; ++j) {
        x[j] = row[j * 32 + lane];
        s += x[j]; sq += x[j] * x[j];
    }
    #pragma unroll
    for (int off = 16; off > 0; off >>= 1) {
        s  += __shfl_xor(s,  off, 32);
        sq += __shfl_xor(sq, off, 32);
    }
    float mu  = s / (float)DD;
    float var = sq / (float)DD - mu * mu;
    float rs  = rsqrtf(var + 1e-5f);
    const float* g = (node < NI) ? g1 : g2;
    const float* be = (node < NI) ? b1 : b2;
    #pragma unroll
    for (int j = 0; j < DD / 32; ++j) {
        int o = j * 32 + lane;
        row[o] = (x[j] - mu) * rs * g[o] + be[o];
    }
}

extern "C" void kernel_launch(void* const* d_in, const int* in_sizes, int n_in,
                              void* d_out, int out_size, void* d_ws, size_t ws_size,
                              hipStream_t stream) {
    const float* feat = (const float*)d_in[0];
    const int*   E    = (const int*)d_in[1];
    const float* Wi   = (const float*)d_in[2];
    const float* ali  = (const float*)d_in[3];
    const float* ari  = (const float*)d_in[4];
    const float* bi   = (const float*)d_in[5];
    const float* Wpi  = (const float*)d_in[6];
    const float* bpi  = (const float*)d_in[7];
    const float* g1   = (const float*)d_in[8];
    const float* be1  = (const float*)d_in[9];
    const float* Wa   = (const float*)d_in[10];
    const float* ala  = (const float*)d_in[11];
    const float* ara  = (const float*)d_in[12];
    const float* ba   = (const float*)d_in[13];
    const float* Wpa  = (const float*)d_in[14];
    const float* bpa  = (const float*)d_in[15];
    const float* g2   = (const float*)d_in[16];
    const float* be2  = (const float*)d_in[17];
    float* out = (float*)d_out;

    char* ws = (char*)d_ws;
    size_t o_i2a   = 0;
    size_t o_a2i   = o_i2a   + (size_t)BB * NI * 8;
    size_t o_colT  = o_a2i   + (size_t)BB * NA * 16 * 4;
    size_t o_mimgT = o_colT  + (size_t)BB * NA * 16 * 4;
    size_t o_maudT = o_mimgT + (size_t)BB * NI * 16 * 4;
    size_t o_wh    = o_maudT + (size_t)BB * NA * 8;
    size_t o_el    = o_wh    + (size_t)BB * NT * DD * 4;
    size_t o_er    = o_el    + (size_t)BB * NT * 4;
    size_t o_gat   = o_er    + (size_t)BB * NT * 4;

    unsigned* i2a_row  = (unsigned*)(ws + o_i2a);
    unsigned* a2i_row  = (unsigned*)(ws + o_a2i);
    unsigned* i2a_colT = (unsigned*)(ws + o_colT);
    unsigned* mimgT    = (unsigned*)(ws + o_mimgT);
    unsigned* maudT    = (unsigned*)(ws + o_maudT);
    float*    wh       = (float*)(ws + o_wh);
    float*    el       = (float*)(ws + o_el);
    float*    er       = (float*)(ws + o_er);
    float*    gat      = (float*)(ws + o_gat);

    // 1) zero adjacency accumulators (I2A_row + A2I_row + I2A_colT)
    int zwords = (int)((o_mimgT - o_i2a) / 4);
    k_zero<<<(zwords + 255) / 256, 256, 0, stream>>>((unsigned*)ws, zwords);

    // 2) build bitmask adjacency
    k_build_adj<<<BB, 256, 0, stream>>>(E, i2a_row, a2i_row, i2a_colT);

    // 3) metapath masks
    k_mimgT<<<(BB * NI * 16) / 256, 256, 0, stream>>>(a2i_row, i2a_colT, mimgT);
    k_maudT<<<(BB * NA) / 256, 256, 0, stream>>>(a2i_row, i2a_colT, maudT);

    // 4) Wh = h @ W^T for both subgraphs (fp32 WMMA)
    k_gemm_wmma<<<(BB * 32 * 12) / 8, 256, 0, stream>>>(feat, Wi, nullptr, wh, 0,  32);
    k_gemm_wmma<<<(BB * 3  * 12) / 8, 256, 0, stream>>>(feat, Wa, nullptr, wh, NI, 3);

    // 5) attention logits
    k_elr<<<(BB * NT) / 256, 256, 0, stream>>>(wh, ali, ari, ala, ara, el, er);

    // 6) masked softmax aggregation + bias + elu
    k_gat<<<(BB * NI) / 8, 256, 0, stream>>>(mimgT, 16, 16, wh, el, er, bi, gat, 0,  NI);
    k_gat<<<(BB * NA) / 8, 256, 0, stream>>>(maudT,  2,  2, wh, el, er, ba, gat, NI, NA);

    // 7) projection GEMM + bias directly into output (fp32 WMMA)
    k_gemm_wmma<<<(BB * 32 * 12) / 8, 256, 0, stream>>>(gat, Wpi, bpi, out, 0,  32);
    k_gemm_wmma<<<(BB * 3  * 12) / 8, 256, 0, stream>>>(gat, Wpa, bpa, out, NI, 3);

    // 8) layernorm in place
    k_layernorm<<<(BB * NT) / 8, 256, 0, stream>>>(out, g1, be1, g2, be2);
}